// TrivialModel_38517266711057
// MI455X (gfx1250) — compile-verified
//
#include <hip/hip_runtime.h>

typedef __attribute__((ext_vector_type(2))) float v2f;
typedef __attribute__((ext_vector_type(8))) float v8f;

#define B_DIM 32
#define T_DIM 12
#define N_DIM 100000
#define HZ    10
#define R_DIM 64

// One block = 256 threads = 8 waves (wave32). Each wave produces means for 32
// consecutive nodes of one batch via two 16-node WMMA tiles.
//
// WMMA mapping (V_WMMA_F32_16X16X4_F32, D = A*B + C):
//   A[m,k] = x[b, t_base+k, n0+m, 0]   (documented layout: lane=M, VGPR0=K{0,2}, VGPR1=K{1,3})
//   B[k,n] = 1/12 (constant -> B layout irrelevant)
//   => D[m,n] = (1/12) * sum_k x[b, t_base+k, n0+m, 0]  for every column n.
// Accumulating over t_base = 0,4,8 gives the T=12 mean in fp32.
__global__ __launch_bounds__(256) void mean_wmma_kernel(
    const float* __restrict__ x, const int* __restrict__ cid,
    float* __restrict__ pred, float* __restrict__ seg, float* __restrict__ cnt)
{
  __shared__ float lmean[256];   // per-wave transposed means (8 waves * 32 nodes)
  __shared__ float lsum[R_DIM];  // regional partial sums for this block
  __shared__ float lcnt[R_DIM];  // regional counts (b==0 blocks only)

  const int tid = threadIdx.x;
  const int b   = blockIdx.y;
  if (tid < R_DIM) { lsum[tid] = 0.0f; lcnt[tid] = 0.0f; }
  __syncthreads();

  const int wave = tid >> 5;
  const int lane = tid & 31;
  const int half = lane >> 4;   // 0: K slots {0,1}; 1: K slots {2,3}
  const int m    = lane & 15;   // A-matrix row (node within tile)
  const int n0   = blockIdx.x * 256 + wave * 32;

  v2f bw; bw.x = 1.0f / 12.0f; bw.y = 1.0f / 12.0f;   // constant B matrix

  v8f d0 = {0.f,0.f,0.f,0.f,0.f,0.f,0.f,0.f};
  v8f d1 = {0.f,0.f,0.f,0.f,0.f,0.f,0.f,0.f};

  const size_t xb  = (size_t)b * (size_t)(T_DIM * 2) * (size_t)N_DIM; // batch base (elems)
  const size_t trs = (size_t)N_DIM * 2u;                              // one time-step stride
  const int na = min(n0 + m,      N_DIM - 1);   // tile 0 node (clamped, select not branch)
  const int nb = min(n0 + 16 + m, N_DIM - 1);   // tile 1 node

  #pragma unroll
  for (int g = 0; g < 3; ++g) {
    const int t0 = 4 * g + 2 * half;           // time for K slot 0; slot 1 is t0+1
    const size_t row = xb + (size_t)t0 * trs;
    v2f a0, a1;
    a0.x = x[row       + (size_t)na * 2u];
    a0.y = x[row + trs + (size_t)na * 2u];
    a1.x = x[row       + (size_t)nb * 2u];
    a1.y = x[row + trs + (size_t)nb * 2u];
    d0 = __builtin_amdgcn_wmma_f32_16x16x4_f32(false, a0, false, bw, (short)0, d0, false, false);
    d1 = __builtin_amdgcn_wmma_f32_16x16x4_f32(false, a1, false, bw, (short)0, d1, false, false);
  }

  // D layout: VGPR j -> M=j (lanes 0-15) / M=8+j (lanes 16-31); all columns equal.
  // Lane 0 holds means for nodes n0+0..7, lane 16 for n0+8..15 (tile 0); same for tile 1.
  if ((lane & 15) == 0) {
    const int base = wave * 32 + half * 8;
    #pragma unroll
    for (int j = 0; j < 8; ++j) lmean[base + j]      = d0[j];
    #pragma unroll
    for (int j = 0; j < 8; ++j) lmean[base + 16 + j] = d1[j];
  }
  __syncthreads();

  const int   n    = n0 + lane;
  const float mean = lmean[wave * 32 + lane];
  if (n < N_DIM) {
    // 10 horizon copies: 32 consecutive floats per wave per store -> fully coalesced.
    size_t p = (size_t)b * (size_t)(HZ * (size_t)N_DIM) + (size_t)n;
    #pragma unroll
    for (int h = 0; h < HZ; ++h) pred[p + (size_t)h * N_DIM] = mean;

    const int r = cid[n] & (R_DIM - 1);
    atomicAdd(&lsum[r], mean);                 // LDS float atomic (ds_add)
    if (b == 0) atomicAdd(&lcnt[r], 1.0f);
  }
  __syncthreads();

  if (tid < R_DIM) {
    unsafeAtomicAdd(&seg[b * R_DIM + tid], lsum[tid]);   // global_atomic_add_f32
    if (b == 0) unsafeAtomicAdd(&cnt[tid], lcnt[tid]);
  }
}

__global__ void finalize_regional(const float* __restrict__ seg,
                                  const float* __restrict__ cnt,
                                  float* __restrict__ out2)
{
  const int i = blockIdx.x * blockDim.x + threadIdx.x;   // i = (b*10+h)*64 + r
  if (i >= B_DIM * HZ * R_DIM) return;
  const int r  = i & (R_DIM - 1);
  const int bh = i >> 6;
  const int b  = bh / HZ;
  out2[i] = seg[b * R_DIM + r] / cnt[r];
}

extern "C" void kernel_launch(void* const* d_in, const int* in_sizes, int n_in,
                              void* d_out, int out_size, void* d_ws, size_t ws_size,
                              hipStream_t stream)
{
  (void)in_sizes; (void)n_in; (void)out_size; (void)ws_size;

  const float* x   = (const float*)d_in[0];
  const int*   cid = (const int*)d_in[1];
  float* out = (float*)d_out;
  float* seg = (float*)d_ws;              // [B_DIM * R_DIM]
  float* cnt = seg + B_DIM * R_DIM;       // [R_DIM]

  // Re-zero accumulators every call (graph-capture-safe memset node).
  hipMemsetAsync(d_ws, 0, (size_t)(B_DIM * R_DIM + R_DIM) * sizeof(float), stream);

  dim3 grid((N_DIM + 255) / 256, B_DIM, 1);
  mean_wmma_kernel<<<grid, 256, 0, stream>>>(x, cid, out, seg, cnt);

  const int tot = B_DIM * HZ * R_DIM;
  finalize_regional<<<(tot + 255) / 256, 256, 0, stream>>>(
      seg, cnt, out + (size_t)B_DIM * HZ * N_DIM);
}